// GLNID_11905649344752
// MI455X (gfx1250) — compile-verified
//
#include <hip/hip_runtime.h>

// ---------------------------------------------------------------------------
// Problem constants (from the reference)
// ---------------------------------------------------------------------------
#define EMB   512
#define NROWS 65536
#define ROWS_PER_BLOCK 256
#define NBLK  (NROWS / ROWS_PER_BLOCK)   // 256 pass-1 blocks per stream

typedef __attribute__((ext_vector_type(2))) float v2f;
typedef __attribute__((ext_vector_type(8))) float v8f;

// ---------------------------------------------------------------------------
// Workspace layout (float offsets). Total ~3.03 MB.
// ---------------------------------------------------------------------------
#define WS_VECS   0                       // 5*512 : r1, r2, u1, u2, edge
#define WS_OUTS   (5*EMB)                 // 5*512 : s1, s2, t1, t2, new(lrelu'd)
#define WS_EW     (10*EMB)                // 4*512 : edge*w_uu, edge*w_ui, edge*w_iu, edge*w_ii
#define WS_A1     (14*EMB)                // 65536
#define WS_A2     (WS_A1 + NROWS)
#define WS_B1     (WS_A2 + NROWS)
#define WS_B2     (WS_B1 + NROWS)
#define WS_P_R1   (WS_B2 + NROWS)         // NBLK*512 partial sums
#define WS_P_R2   (WS_P_R1 + NBLK*EMB)
#define WS_P_U1   (WS_P_R2 + NBLK*EMB)
#define WS_P_U2   (WS_P_U1 + NBLK*EMB)

__device__ __forceinline__ float lrelu(float z) { return z > 0.f ? z : 0.01f * z; }

// ---------------------------------------------------------------------------
// Kernel 0: edge vector + fused (edge ⊙ w) dot-weight vectors
// ---------------------------------------------------------------------------
__global__ void k_prep(const float* __restrict__ buy, const int* __restrict__ edge_nb,
                       const float* __restrict__ w_uu, const float* __restrict__ w_ui,
                       const float* __restrict__ w_iu, const float* __restrict__ w_ii,
                       float* __restrict__ ws) {
  int c = threadIdx.x + blockIdx.x * blockDim.x;
  if (c >= EMB) return;
  float e = buy[(size_t)edge_nb[0] * EMB + c] * 0.1f;
  ws[WS_VECS + 4 * EMB + c] = e;            // edge (scaled)
  ws[WS_EW + 0 * EMB + c] = e * w_uu[c];
  ws[WS_EW + 1 * EMB + c] = e * w_ui[c];
  ws[WS_EW + 2 * EMB + c] = e * w_iu[c];
  ws[WS_EW + 3 * EMB + c] = e * w_ii[c];
}

// ---------------------------------------------------------------------------
// Kernel 1 (per stream): gather rows, 3 row-dots, 2 weighted column reductions.
// One wave per row at a time (lane owns 16 columns); deterministic fixed-order
// reductions (butterfly shuffle in-wave, per-wave LDS slices reduced in order).
// ---------------------------------------------------------------------------
__global__ void k_pass1(const float* __restrict__ emb, const int* __restrict__ nb,
                        const float* __restrict__ w0, const float* __restrict__ w1,
                        const float* __restrict__ w2,
                        float* __restrict__ dot0, float* __restrict__ dot1,
                        float* __restrict__ part0, float* __restrict__ part1) {
  __shared__ float sW[3 * EMB];
  __shared__ float sAcc0[8][EMB];
  __shared__ float sAcc1[8][EMB];

  for (int c = threadIdx.x; c < EMB; c += blockDim.x) {
    sW[c] = w0[c]; sW[EMB + c] = w1[c]; sW[2 * EMB + c] = w2[c];
  }
  __syncthreads();

  const int wave = threadIdx.x >> 5;
  const int lane = threadIdx.x & 31;
  const int colBase = lane * 16;

  float acc0[16], acc1[16];
  #pragma unroll
  for (int j = 0; j < 16; ++j) { acc0[j] = 0.f; acc1[j] = 0.f; }

  const int rowBase = blockIdx.x * ROWS_PER_BLOCK + wave * 32;
  for (int r = 0; r < 32; ++r) {
    const int row = rowBase + r;
    const int idx = nb[row];
    const float4* rp = (const float4*)(emb + (size_t)idx * EMB + colBase);
    float xv[16];
    #pragma unroll
    for (int q = 0; q < 4; ++q) {
      float4 v = rp[q];
      xv[q * 4 + 0] = v.x * 0.1f; xv[q * 4 + 1] = v.y * 0.1f;
      xv[q * 4 + 2] = v.z * 0.1f; xv[q * 4 + 3] = v.w * 0.1f;
    }
    float d0 = 0.f, d1 = 0.f, d2 = 0.f;
    #pragma unroll
    for (int j = 0; j < 16; ++j) {
      d0 += xv[j] * sW[colBase + j];
      d1 += xv[j] * sW[EMB + colBase + j];
      d2 += xv[j] * sW[2 * EMB + colBase + j];
    }
    #pragma unroll
    for (int m = 16; m >= 1; m >>= 1) {      // butterfly: all lanes get full dot
      d0 += __shfl_xor(d0, m, 32);
      d1 += __shfl_xor(d1, m, 32);
      d2 += __shfl_xor(d2, m, 32);
    }
    if (lane == 0) { dot0[row] = d0; dot1[row] = d1; }
    #pragma unroll
    for (int j = 0; j < 16; ++j) { acc0[j] += d0 * xv[j]; acc1[j] += d2 * xv[j]; }
  }

  #pragma unroll
  for (int j = 0; j < 16; ++j) {
    sAcc0[wave][colBase + j] = acc0[j];
    sAcc1[wave][colBase + j] = acc1[j];
  }
  __syncthreads();

  for (int c = threadIdx.x; c < EMB; c += blockDim.x) {
    float s0 = 0.f, s1 = 0.f;
    #pragma unroll
    for (int w = 0; w < 8; ++w) { s0 += sAcc0[w][c]; s1 += sAcc1[w][c]; }
    part0[(size_t)blockIdx.x * EMB + c] = s0;
    part1[(size_t)blockIdx.x * EMB + c] = s1;
  }
}

// ---------------------------------------------------------------------------
// Kernel 2: fixed-order cross-block reduction of the 4 partial vectors
// ---------------------------------------------------------------------------
__global__ void k_reduce(float* __restrict__ ws) {
  int gid = blockIdx.x * blockDim.x + threadIdx.x;
  if (gid >= 4 * EMB) return;
  const int v = gid >> 9;
  const int c = gid & (EMB - 1);
  const int srcOff[4] = { WS_P_R1, WS_P_R2, WS_P_U1, WS_P_U2 };
  const float* p = ws + srcOff[v] + c;
  float s = 0.f;
  for (int b = 0; b < NBLK; ++b) s += p[(size_t)b * EMB];
  ws[WS_VECS + v * EMB + c] = s;            // r1, r2, u1, u2
}

// ---------------------------------------------------------------------------
// Kernel 3: five 512x512 GEMVs via V_WMMA_F32_16X16X4_F32.
// A (16x4) = vector chunk replicated across the 16 M rows:
//   VGPR0: K=0 (lanes 0-15) / K=2 (lanes 16-31); VGPR1: K=1 / K=3.
// B (4x16) lane-striped like C/D: VGPR0: K=0 / K=2 ; VGPR1: K=1 / K=3, N=lane&15.
// D rows all identical; row M=0 lives in VGPR0 lanes 0-15.
// One wave per (matrix, 16-col tile): 5*32 = 160 waves, 128 WMMAs each.
// ---------------------------------------------------------------------------
__global__ void k_gemv_wmma(const float* __restrict__ vecs,   // ws + WS_VECS
                            float* __restrict__ outs,         // ws + WS_OUTS
                            const float* __restrict__ W0, const float* __restrict__ W1,
                            const float* __restrict__ W2, const float* __restrict__ W3,
                            const float* __restrict__ W4) {
  const int m    = blockIdx.x >> 5;    // which (vector, matrix) pair
  const int tile = blockIdx.x & 31;    // 16-column tile
  const float* W = (m == 0) ? W0 : (m == 1) ? W1 : (m == 2) ? W2 : (m == 3) ? W3 : W4;
  const float* vec = vecs + m * EMB;

  const int lane = threadIdx.x;        // blockDim.x == 32, EXEC all ones
  const int ko   = (lane >= 16) ? 2 : 0;
  const int nc   = tile * 16 + (lane & 15);

  v8f acc = { 0.f, 0.f, 0.f, 0.f, 0.f, 0.f, 0.f, 0.f };
  for (int k = 0; k < EMB; k += 4) {
    v2f a, b;
    a.x = vec[k + ko];
    a.y = vec[k + ko + 1];
    b.x = W[(size_t)(k + ko) * EMB + nc];
    b.y = W[(size_t)(k + ko + 1) * EMB + nc];
    acc = __builtin_amdgcn_wmma_f32_16x16x4_f32(false, a, false, b,
                                                (short)0, acc, false, false);
  }
  if (lane < 16) {
    float r = acc[0];                  // M=0 row, N = tile*16 + lane
    if (m == 4) r = lrelu(r);          // new = lrelu(edge @ W_e)
    outs[m * EMB + tile * 16 + lane] = r;
  }
}

// ---------------------------------------------------------------------------
// Kernel 4 (per stream): closed-form output pass.
// out[i,j] = 0.5*(0.5*(lrelu(dA[i]*sA[j]) + lrelu(dB[i]*sB[j])) + 0.1*emb[nb[i]][j])
// Block of 256 threads handles 2 rows (128 threads x float4 per row).
// ---------------------------------------------------------------------------
__global__ void k_final(const float* __restrict__ emb, const int* __restrict__ nb,
                        const float* __restrict__ dA, const float* __restrict__ dB,
                        const float* __restrict__ sA, const float* __restrict__ sB,
                        float* __restrict__ out) {
  const int row = blockIdx.x * 2 + (threadIdx.x >> 7);
  const int c   = (threadIdx.x & 127) * 4;
  const int idx = nb[row];
  const float a = dA[row];
  const float b = dB[row];
  const float4 xv = *(const float4*)(emb + (size_t)idx * EMB + c);
  const float4 s1 = *(const float4*)(sA + c);
  const float4 s2 = *(const float4*)(sB + c);
  float4 o;
  o.x = 0.5f * (0.5f * (lrelu(a * s1.x) + lrelu(b * s2.x)) + 0.1f * xv.x);
  o.y = 0.5f * (0.5f * (lrelu(a * s1.y) + lrelu(b * s2.y)) + 0.1f * xv.y);
  o.z = 0.5f * (0.5f * (lrelu(a * s1.z) + lrelu(b * s2.z)) + 0.1f * xv.z);
  o.w = 0.5f * (0.5f * (lrelu(a * s1.w) + lrelu(b * s2.w)) + 0.1f * xv.w);
  *(float4*)(out + (size_t)row * EMB + c) = o;
}

// ---------------------------------------------------------------------------
// Kernel 5: e_out = 0.5*(new + edge)
// ---------------------------------------------------------------------------
__global__ void k_eout(const float* __restrict__ ws, float* __restrict__ out) {
  int c = threadIdx.x + blockIdx.x * blockDim.x;
  if (c < EMB) out[c] = 0.5f * (ws[WS_OUTS + 4 * EMB + c] + ws[WS_VECS + 4 * EMB + c]);
}

// ---------------------------------------------------------------------------
extern "C" void kernel_launch(void* const* d_in, const int* in_sizes, int n_in,
                              void* d_out, int out_size, void* d_ws, size_t ws_size,
                              hipStream_t stream) {
  (void)in_sizes; (void)n_in; (void)out_size; (void)ws_size;
  const float* user_emb = (const float*)d_in[0];
  const float* item_emb = (const float*)d_in[1];
  const float* buy      = (const float*)d_in[2];
  const float* w_uu     = (const float*)d_in[3];
  const float* w_ui     = (const float*)d_in[4];
  const float* w_iu     = (const float*)d_in[5];
  const float* w_ii     = (const float*)d_in[6];
  const float* W_e      = (const float*)d_in[7];
  const float* W_uu1    = (const float*)d_in[8];
  const float* W_ui1    = (const float*)d_in[9];
  const float* W_iu1    = (const float*)d_in[10];
  const float* W_ii1    = (const float*)d_in[11];
  const int*   user_nb  = (const int*)d_in[12];
  const int*   item_nb  = (const int*)d_in[13];
  const int*   edge_nb  = (const int*)d_in[14];
  float* out = (float*)d_out;
  float* ws  = (float*)d_ws;

  k_prep<<<2, 256, 0, stream>>>(buy, edge_nb, w_uu, w_ui, w_iu, w_ii, ws);

  // user stream: a1 = x·(e⊙w_uu), a2 = x·(e⊙w_ui), b3 = x·(e⊙w_iu); r1 += a1*x, u2 += b3*x
  k_pass1<<<NBLK, 256, 0, stream>>>(user_emb, user_nb,
      ws + WS_EW + 0 * EMB, ws + WS_EW + 1 * EMB, ws + WS_EW + 2 * EMB,
      ws + WS_A1, ws + WS_A2, ws + WS_P_R1, ws + WS_P_U2);

  // item stream: b1 = y·(e⊙w_ii), b2 = y·(e⊙w_iu), a3 = y·(e⊙w_ui); u1 += b1*y, r2 += a3*y
  k_pass1<<<NBLK, 256, 0, stream>>>(item_emb, item_nb,
      ws + WS_EW + 3 * EMB, ws + WS_EW + 2 * EMB, ws + WS_EW + 1 * EMB,
      ws + WS_B1, ws + WS_B2, ws + WS_P_U1, ws + WS_P_R2);

  k_reduce<<<8, 256, 0, stream>>>(ws);

  // s1 = r1@W_uu1, s2 = r2@W_ui1, t1 = u1@W_ii1, t2 = u2@W_iu1, new = lrelu(edge@W_e)
  k_gemv_wmma<<<160, 32, 0, stream>>>(ws + WS_VECS, ws + WS_OUTS,
                                      W_uu1, W_ui1, W_ii1, W_iu1, W_e);

  k_final<<<NROWS / 2, 256, 0, stream>>>(user_emb, user_nb,
      ws + WS_A1, ws + WS_A2, ws + WS_OUTS + 0 * EMB, ws + WS_OUTS + 1 * EMB, out);

  k_final<<<NROWS / 2, 256, 0, stream>>>(item_emb, item_nb,
      ws + WS_B1, ws + WS_B2, ws + WS_OUTS + 2 * EMB, ws + WS_OUTS + 3 * EMB,
      out + (size_t)NROWS * EMB);

  k_eout<<<2, 256, 0, stream>>>(ws, out + 2 * (size_t)NROWS * EMB);
}